// F2VConv3d_10763188043852
// MI455X (gfx1250) — compile-verified
//
#include <hip/hip_runtime.h>

// ---------------------------------------------------------------------------
// F2VConv3d for MI455X (gfx1250, wave32).
// Memory-bound pipeline (~300MB @ 23.3TB/s ~ 15us); keep f32 end-to-end and
// use V_WMMA_F32_16X16X4_F32 for the [200k x 64] x [64 x 128] pointwise GEMM.
// ---------------------------------------------------------------------------

typedef __attribute__((ext_vector_type(2))) float v2f;
typedef __attribute__((ext_vector_type(8))) float v8f;

#define NFACE 400000
#define NVERT 200000
#define C_IN 64
#define C_OUT 128
#define KFILT 27
#define BN_EPS 1e-5f

// ---------------------------------------------------------------------------
// k0: zero agg [NVERT*C_IN] and stats [2*C_OUT] in workspace
// ---------------------------------------------------------------------------
__global__ __launch_bounds__(256) void k0_zero(float* __restrict__ buf, int n) {
    int i = blockIdx.x * 256 + threadIdx.x;
    if (i < n) buf[i] = 0.0f;
}

// ---------------------------------------------------------------------------
// k1: one wave per face. Lane handles channels {lane, lane+32}.
//   w[c]    = sum_k filt_coeff[f][k] * SW[k][c]   (SW staged in LDS, 6.75KB)
//   contrib = inputs[f][c] * w[c]
//   atomic scatter-add to the 3 vertices (agg fits in L2 -> fast atomics)
// grid = NFACE/8 blocks x 256 threads (8 waves), exact.
// ---------------------------------------------------------------------------
__global__ __launch_bounds__(256) void k1_face_scatter(
    const float* __restrict__ inputs,      // [NFACE, C_IN]
    const float* __restrict__ filt_coeff,  // [NFACE, KFILT]
    const int*   __restrict__ face,        // [NFACE, 3]
    const float* __restrict__ SW,          // [KFILT, C_IN] (mult==1)
    float*       __restrict__ agg)         // [NVERT, C_IN]
{
    __shared__ float sSW[KFILT * C_IN];
    for (int i = threadIdx.x; i < KFILT * C_IN; i += 256) sSW[i] = SW[i];
    __syncthreads();

    int f    = blockIdx.x * 8 + (threadIdx.x >> 5);
    int lane = threadIdx.x & 31;

    const float* fcf = filt_coeff + (size_t)f * KFILT;
    float w0 = 0.0f, w1 = 0.0f;
#pragma unroll
    for (int k = 0; k < KFILT; ++k) {
        float ck = fcf[k];  // uniform across wave -> broadcast
        w0 = fmaf(ck, sSW[k * C_IN + lane],      w0);
        w1 = fmaf(ck, sSW[k * C_IN + lane + 32], w1);
    }
    float c0 = inputs[(size_t)f * C_IN + lane]      * w0;
    float c1 = inputs[(size_t)f * C_IN + lane + 32] * w1;

    int v0 = face[f * 3 + 0], v1 = face[f * 3 + 1], v2 = face[f * 3 + 2];
    atomicAdd(&agg[(size_t)v0 * C_IN + lane],      c0);
    atomicAdd(&agg[(size_t)v0 * C_IN + lane + 32], c1);
    atomicAdd(&agg[(size_t)v1 * C_IN + lane],      c0);
    atomicAdd(&agg[(size_t)v1 * C_IN + lane + 32], c1);
    atomicAdd(&agg[(size_t)v2 * C_IN + lane],      c0);
    atomicAdd(&agg[(size_t)v2 * C_IN + lane + 32], c1);
}

// ---------------------------------------------------------------------------
// k2: WMMA GEMM  out[m, n] = relu( (agg[vt_map[m]]/denom) @ DW + bias )
// One 16x16 output tile per wave; 8 waves/block cover all 8 N-tiles of one
// 16-row M-tile. DW (64x128 f32 = 32KB) staged in LDS once per block.
// K=64 consumed in 16 steps of V_WMMA_F32_16X16X4_F32.
//
// f32 WMMA VGPR layouts (ISA 7.12.2):
//   A 16x4 : lane lh=lane&15 owns row M=m0+lh; VGPR0/1 hold K=klo/klo+1,
//            klo = 0 (lanes 0-15) or 2 (lanes 16-31).
//   B 4x16 : lane lh owns col N=n0+lh; VGPR0/1 hold K=klo/klo+1 (mirrored).
//   C/D    : VGPR r = row m0 + r + (lane<16 ? 0 : 8), col n0+lh.
// Also accumulates per-channel sum/sum^2 for BN via L2 atomics.
// grid = NVERT/16 = 12500 blocks, exact -> EXEC all-ones (WMMA requirement).
// ---------------------------------------------------------------------------
__global__ __launch_bounds__(256) void k2_gemm_bias_relu(
    const float*     __restrict__ agg,       // [NVERT, C_IN]
    const int*       __restrict__ nf_count,  // [NVERT]
    const long long* __restrict__ vt_map,    // [NVERT]
    const float*     __restrict__ DW,        // [C_IN, C_OUT]
    const float*     __restrict__ bias,      // [1, C_OUT]
    float*           __restrict__ out,       // [NVERT, C_OUT] (pre-BN)
    float*           __restrict__ stats)     // [2*C_OUT]: sums, sumsqs
{
    __shared__ float sDW[C_IN * C_OUT];  // 32 KB of the 320KB WGP LDS
    for (int i = threadIdx.x; i < C_IN * C_OUT; i += 256) sDW[i] = DW[i];
    __syncthreads();

    const int warp = threadIdx.x >> 5;
    const int lane = threadIdx.x & 31;
    const int lh   = lane & 15;
    const int klo  = (lane < 16) ? 0 : 2;
    const int m0   = blockIdx.x * 16;
    const int n0   = warp * 16;

    // Fuse vt_map gather + 1/nf_count scaling into the A-matrix load.
    long long vr = vt_map[m0 + lh];
    int nc = nf_count[vr];
    float rdenom = 1.0f / (float)(nc > 1 ? nc : 1);
    const float* aRow = agg + (size_t)vr * C_IN;

    v8f c = {};
#pragma unroll
    for (int k0 = 0; k0 < C_IN; k0 += 4) {
        v2f a;
        a.x = aRow[k0 + klo]     * rdenom;
        a.y = aRow[k0 + klo + 1] * rdenom;
        v2f b;
        b.x = sDW[(k0 + klo)     * C_OUT + n0 + lh];
        b.y = sDW[(k0 + klo + 1) * C_OUT + n0 + lh];
        // (neg_a, A, neg_b, B, c_mod, C, reuse_a, reuse_b)
        c = __builtin_amdgcn_wmma_f32_16x16x4_f32(
                false, a, false, b, (short)0, c, false, false);
    }

    // Epilogue: bias + ReLU, store pre-BN result, accumulate BN statistics.
    const int col   = n0 + lh;
    const int mbase = m0 + ((lane < 16) ? 0 : 8);
    const float bval = bias[col];
    float bsum = 0.0f, bsq = 0.0f;
#pragma unroll
    for (int r = 0; r < 8; ++r) {
        float v = c[r] + bval;
        v = v > 0.0f ? v : 0.0f;
        out[(size_t)(mbase + r) * C_OUT + col] = v;
        bsum += v;
        bsq  += v * v;
    }
    atomicAdd(&stats[col],         bsum);
    atomicAdd(&stats[C_OUT + col], bsq);
}

// ---------------------------------------------------------------------------
// k3: in-place BatchNorm on d_out using accumulated stats.
// mean = sum/NV ; var = sumsq/NV - mean^2 (biased, matches jnp.var)
// grid = NVERT*C_OUT/256 = 100000 blocks, exact.
// ---------------------------------------------------------------------------
__global__ __launch_bounds__(256) void k3_batchnorm(
    float*       __restrict__ out,
    const float* __restrict__ stats,
    const float* __restrict__ gamma,
    const float* __restrict__ beta)
{
    size_t i = (size_t)blockIdx.x * 256 + threadIdx.x;
    int col = (int)(i & (C_OUT - 1));
    float mean = stats[col] * (1.0f / (float)NVERT);
    float var  = stats[C_OUT + col] * (1.0f / (float)NVERT) - mean * mean;
    float scale = rsqrtf(var + BN_EPS) * gamma[col];
    out[i] = (out[i] - mean) * scale + beta[col];
}

// ---------------------------------------------------------------------------
// launch
// ---------------------------------------------------------------------------
extern "C" void kernel_launch(void* const* d_in, const int* in_sizes, int n_in,
                              void* d_out, int out_size, void* d_ws, size_t ws_size,
                              hipStream_t stream) {
    const float*     inputs     = (const float*)d_in[0];
    const float*     filt_coeff = (const float*)d_in[1];
    const int*       face       = (const int*)d_in[2];
    const int*       nf_count   = (const int*)d_in[3];
    const long long* vt_map     = (const long long*)d_in[4];
    const float*     SW         = (const float*)d_in[5];
    const float*     DW         = (const float*)d_in[6];
    const float*     biases     = (const float*)d_in[7];
    const float*     gamma      = (const float*)d_in[8];
    const float*     beta       = (const float*)d_in[9];
    float* out = (float*)d_out;

    // workspace layout: agg [NVERT*C_IN] | stats [2*C_OUT]
    float* agg   = (float*)d_ws;
    float* stats = agg + (size_t)NVERT * C_IN;
    int nzero = NVERT * C_IN + 2 * C_OUT;

    k0_zero<<<(nzero + 255) / 256, 256, 0, stream>>>(agg, nzero);
    k1_face_scatter<<<NFACE / 8, 256, 0, stream>>>(inputs, filt_coeff, face, SW, agg);
    k2_gemm_bias_relu<<<NVERT / 16, 256, 0, stream>>>(agg, nf_count, vt_map, DW,
                                                      biases, out, stats);
    k3_batchnorm<<<(NVERT * C_OUT) / 256, 256, 0, stream>>>(out, stats, gamma, beta);
}